// SoftQuantilesLayer_51427938402320
// MI455X (gfx1250) — compile-verified
//
#include <hip/hip_runtime.h>
#include <hip/hip_bf16.h>
#include <math.h>

typedef __attribute__((ext_vector_type(2))) float v2f;
typedef __attribute__((ext_vector_type(8))) float v8f;

#define N_SAMP   8192
#define M_TGT    7
#define TPB      512
#define NWAVES   16          // TPB / 32
#define EPT      16          // elements per thread
#define N_ITERS  100

#define LOG2E_F  1.4426950408889634f
#define LN2_F    0.6931471805599453f

__device__ __forceinline__ float wave_sum(float v) {
#pragma unroll
  for (int m = 16; m >= 1; m >>= 1) v += __shfl_xor(v, m, 32);
  return v;
}
__device__ __forceinline__ float wave_max(float v) {
#pragma unroll
  for (int m = 16; m >= 1; m >>= 1) v = fmaxf(v, __shfl_xor(v, m, 32));
  return v;
}
__device__ __forceinline__ float wave_min(float v) {
#pragma unroll
  for (int m = 16; m >= 1; m >>= 1) v = fminf(v, __shfl_xor(v, m, 32));
  return v;
}

// Broadcast lane 0's value to the whole wave. EXEC is all-ones at every call
// site, so the "first active lane" is lane 0. Avoids a ds_bpermute round-trip.
__device__ __forceinline__ float bcast_lane0(float v) {
  return __int_as_float(__builtin_amdgcn_readfirstlane(__float_as_int(v)));
}

// Raw hardware transcendentals (v_exp_f32 / v_log_f32, base 2). Safe here:
// exp2 args are <= 0 (max-stabilized), log2 args are >= 1 (sums of exps with
// at least one term equal to 1), so no denorm/edge fixups are needed.
__device__ __forceinline__ float hw_exp2(float x) { return __builtin_amdgcn_exp2f(x); }
__device__ __forceinline__ float hw_log2(float x) { return __builtin_amdgcn_logf(x); }

// Sum 16 wave-partials x 7 targets using four chained V_WMMA_F32_16X16X4_F32
// (A = staged partials, B = ones). Staging tile is TRANSPOSED:
//   stage_T[j * 16 + w] = partial of target j from wave w   (j=7..15 stay 0)
// A 16x4 lane layout (documented): lanes 0-15 hold rows M=0-15 with K={0,1},
// lanes 16-31 hold the same rows with K={2,3}. WMMA q covers waves 4q..4q+3,
// so lane L's two K-values are stage_T[m*16 + k0+4q] and ...+1 -> ONE aligned
// float2 LDS load per WMMA from a single base address (immediate offsets).
__device__ __forceinline__ v8f wmma_reduce16(const float* stage_T, int lane) {
  const int m_row = lane & 15;
  const int k0 = (lane < 16) ? 0 : 2;
  const float* base = stage_T + m_row * 16 + k0;
  v2f b_ones; b_ones.x = 1.0f; b_ones.y = 1.0f;
  v8f acc = {};
#pragma unroll
  for (int q = 0; q < 4; ++q) {        // waves 4q .. 4q+3
    const v2f a = *reinterpret_cast<const v2f*>(base + 4 * q);
    acc = __builtin_amdgcn_wmma_f32_16x16x4_f32(false, a, false, b_ones,
                                                (short)0, acc, false, false);
  }
  return acc;
}

__global__ __launch_bounds__(TPB) void soft_quantiles_kernel(
    const float* __restrict__ x, float* __restrict__ out) {
  const int row  = blockIdx.x;
  const int tid  = threadIdx.x;
  const int lane = tid & 31;
  const int wave = tid >> 5;

  __shared__ float stage_T[16 * NWAVES];  // [slot j][wave]; j=7..15 stay 0
  __shared__ float lds_small[8];

  if (tid < 16 * NWAVES) stage_T[tid] = 0.0f;
  __syncthreads();

  const float* __restrict__ xrow = x + (size_t)row * N_SAMP;
  __builtin_prefetch(xrow + tid * EPT, 0, 1);  // global_prefetch_b8

  // Thread t owns elements { e4*2048 + t*4 + q : e4 in 0..3, q in 0..3 }.
  float xs[EPT];
#pragma unroll
  for (int e4 = 0; e4 < 4; ++e4) {
    const float4 v = *reinterpret_cast<const float4*>(xrow + e4 * 2048 + tid * 4);
    xs[e4 * 4 + 0] = v.x; xs[e4 * 4 + 1] = v.y;
    xs[e4 * 4 + 2] = v.z; xs[e4 * 4 + 3] = v.w;
  }

  // ---- min-max squash to [0,1]
  float mn = xs[0], mx = xs[0];
#pragma unroll
  for (int i = 1; i < EPT; ++i) { mn = fminf(mn, xs[i]); mx = fmaxf(mx, xs[i]); }
  mn = wave_min(mn);
  mx = wave_max(mx);
  if (lane == 0) { stage_T[0 * 16 + wave] = mn; stage_T[1 * 16 + wave] = mx; }
  __syncthreads();
  if (tid == 0) {
    float a = stage_T[0], b = stage_T[16];
#pragma unroll
    for (int w = 1; w < NWAVES; ++w) {
      a = fminf(a, stage_T[0 * 16 + w]);
      b = fmaxf(b, stage_T[1 * 16 + w]);
    }
    lds_small[0] = a; lds_small[1] = b;
  }
  __syncthreads();
  const float xmin = lds_small[0], xmax = lds_small[1];
  const float range = xmax - xmin;
  const float inv_range = 1.0f / range;
#pragma unroll
  for (int i = 0; i < EPT; ++i) xs[i] = (xs[i] - xmin) * inv_range;

  // ---- target spec (n = 8192, quantiles .25/.5/.75)
  const float yv[M_TGT]   = {0.0f, 1.0f/6.0f, 1.0f/3.0f, 0.5f,
                             2.0f/3.0f, 5.0f/6.0f, 1.0f};
  const float logb[M_TGT] = {-1.38653853f, -9.01091335f, -1.38678276f,
                             -9.01091335f, -1.38678276f, -9.01091335f,
                             -1.38653853f};
  const float LOG_A = -9.01091335f;  // -log(8192)

  float f[EPT];
#pragma unroll
  for (int i = 0; i < EPT; ++i) f[i] = 0.0f;
  float g[M_TGT];
#pragma unroll
  for (int j = 0; j < M_TGT; ++j) g[j] = 0.0f;

  float eps = 0.1f;

#pragma unroll 1
  for (int it = 0; it < N_ITERS; ++it) {
    const float inv_eps_l2 = LOG2E_F / eps;   // fold log2(e) into 1/eps
    const float eps_ln2    = eps * LN2_F;     // eps * ln(2) for log2 -> ln

    // ===== g-update:  g_j = eps*log b_j - M_j - eps*log S_j
    // pass A: per-thread maxes of v_ij = f_i - (xs_i - y_j)^2
    float mj[M_TGT];
#pragma unroll
    for (int j = 0; j < M_TGT; ++j) mj[j] = -3.4e38f;
#pragma unroll
    for (int i = 0; i < EPT; ++i) {
      const float fi = f[i], xi = xs[i];
#pragma unroll
      for (int j = 0; j < M_TGT; ++j) {
        const float t = xi - yv[j];
        mj[j] = fmaxf(mj[j], fi - t * t);
      }
    }
#pragma unroll
    for (int j = 0; j < M_TGT; ++j) mj[j] = wave_max(mj[j]);
    __syncthreads();  // WAR: previous iteration's WMMA reads of stage_T
    if (lane == 0) {
#pragma unroll
      for (int j = 0; j < M_TGT; ++j) stage_T[j * 16 + wave] = mj[j];
    }
    __syncthreads();
    if (tid < M_TGT) {
      float a = stage_T[tid * 16 + 0];
#pragma unroll
      for (int w = 1; w < NWAVES; ++w) a = fmaxf(a, stage_T[tid * 16 + w]);
      lds_small[tid] = a;
    }
    __syncthreads();
    float Mj[M_TGT];
#pragma unroll
    for (int j = 0; j < M_TGT; ++j) Mj[j] = lds_small[j];

    // pass B: stabilized exp-sums, exp2((v - M)/eps * log2e) via v_exp_f32
    float sj[M_TGT];
#pragma unroll
    for (int j = 0; j < M_TGT; ++j) sj[j] = 0.0f;
#pragma unroll
    for (int i = 0; i < EPT; ++i) {
      const float fi = f[i], xi = xs[i];
#pragma unroll
      for (int j = 0; j < M_TGT; ++j) {
        const float t = xi - yv[j];
        sj[j] += hw_exp2((fi - t * t - Mj[j]) * inv_eps_l2);
      }
    }
#pragma unroll
    for (int j = 0; j < M_TGT; ++j) sj[j] = wave_sum(sj[j]);
    if (lane == 0) {
#pragma unroll
      for (int j = 0; j < M_TGT; ++j) stage_T[j * 16 + wave] = sj[j];
    }
    __syncthreads();
    const v8f accS = wmma_reduce16(stage_T, lane);
    // lanes 0-15 hold rows 0..6 in v8f elements 0..6; lane 0 is authoritative
    float Sj[M_TGT];
#pragma unroll
    for (int j = 0; j < M_TGT; ++j) Sj[j] = bcast_lane0(accS[j]);

#pragma unroll
    for (int j = 0; j < M_TGT; ++j)
      g[j] = eps * logb[j] - Mj[j] - eps_ln2 * hw_log2(Sj[j]);

    // ===== f-update (thread-local over the 7 targets):
    //  f_i = eps*log a - N_i - eps*log T_i
#pragma unroll
    for (int i = 0; i < EPT; ++i) {
      const float xi = xs[i];
      float w[M_TGT];
      float nmax = -3.4e38f;
#pragma unroll
      for (int j = 0; j < M_TGT; ++j) {
        const float t = xi - yv[j];
        w[j] = g[j] - t * t;
        nmax = fmaxf(nmax, w[j]);
      }
      float s = 0.0f;
#pragma unroll
      for (int j = 0; j < M_TGT; ++j) s += hw_exp2((w[j] - nmax) * inv_eps_l2);
      f[i] = eps * LOG_A - nmax - eps_ln2 * hw_log2(s);
    }

    eps = fmaxf(eps * 0.95f, 1.0e-3f);
  }

  // ===== final transport plan + barycentric projection (EPS_FINAL = 1e-3)
  const float inv_ef_l2 = 1000.0f * LOG2E_F;  // log2e / 1e-3
  float pj[M_TGT];
#pragma unroll
  for (int j = 0; j < M_TGT; ++j) pj[j] = 0.0f;
#pragma unroll
  for (int i = 0; i < EPT; ++i) {
    const float xsi = xs[i];
    const float xi  = xsi * range + xmin;  // original scale
    const float fi  = f[i];
#pragma unroll
    for (int j = 0; j < M_TGT; ++j) {
      const float t = xsi - yv[j];
      pj[j] += hw_exp2((fi + g[j] - t * t) * inv_ef_l2) * xi;
    }
  }
#pragma unroll
  for (int j = 0; j < M_TGT; ++j) pj[j] = wave_sum(pj[j]);
  __syncthreads();  // WAR vs last iteration's WMMA reads
  if (lane == 0) {
#pragma unroll
    for (int j = 0; j < M_TGT; ++j) stage_T[j * 16 + wave] = pj[j];
  }
  __syncthreads();
  const v8f accP = wmma_reduce16(stage_T, lane);

  if (tid == 0) {
    // lane0 v8f element j == total projection for target j; b[1,3,5] = 1/8192
    out[row * 3 + 0] = accP[1] * 8192.0f;
    out[row * 3 + 1] = accP[3] * 8192.0f;
    out[row * 3 + 2] = accP[5] * 8192.0f;
  }
}

extern "C" void kernel_launch(void* const* d_in, const int* in_sizes, int n_in,
                              void* d_out, int out_size, void* d_ws, size_t ws_size,
                              hipStream_t stream) {
  (void)n_in; (void)out_size; (void)d_ws; (void)ws_size;
  const float* x = (const float*)d_in[0];
  float* out = (float*)d_out;
  const int batch = in_sizes[0] / N_SAMP;  // 256 rows
  soft_quantiles_kernel<<<batch, TPB, 0, stream>>>(x, out);
}